// CCA_SSG_56255481643393
// MI455X (gfx1250) — compile-verified
//
#include <hip/hip_runtime.h>

typedef __attribute__((ext_vector_type(2))) float v2f;
typedef __attribute__((ext_vector_type(8))) float v8f;

#define HID 64

// ---------------- degree / dinv ----------------
__global__ void k_init_ones(float* deg, int n) {
    int i = blockIdx.x * blockDim.x + threadIdx.x;
    if (i < n) deg[i] = 1.0f;  // self-loop
}

__global__ void k_deg_accum(const int* __restrict__ dst, float* __restrict__ deg, int ne) {
    int e = blockIdx.x * blockDim.x + threadIdx.x;
    if (e < ne) atomicAdd(&deg[dst[e]], 1.0f);
}

__global__ void k_rsqrt_ip(float* v, int n) {
    int i = blockIdx.x * blockDim.x + threadIdx.x;
    if (i < n) v[i] = rsqrtf(v[i]);
}

// ---------------- GEMM: H[n x 64] = X[n x K] * W[K x 64] via WMMA f32 16x16x4 ----------------
// block = 128 threads (4 waves). Each wave computes a 16-row x 64-col strip.
template <int K>
__global__ __launch_bounds__(128) void k_gemm_wmma(const float* __restrict__ X,
                                                   const float* __restrict__ W,
                                                   float* __restrict__ H, int nrows) {
    __shared__ float xs[4][16 * 68];  // per-wave 16x64 chunk, row stride 68 (bank-conflict pad)
    const int lane  = threadIdx.x & 31;
    const int wave  = threadIdx.x >> 5;
    const int lrow  = lane & 15;
    const int khalf = lane >> 4;  // 0 or 1

    const int ntiles = nrows >> 4;
    int tile = blockIdx.x * 4 + wave;
    if (tile >= ntiles) tile = ntiles - 1;  // clamp: keep EXEC full & barriers uniform
    const int row0 = tile << 4;

    v8f acc0 = {}, acc1 = {}, acc2 = {}, acc3 = {};

    for (int kc = 0; kc < K; kc += 64) {
        // stage 16 rows x 64 k-values: 256 float4 loads, 8 iterations of 32 lanes (coalesced)
        #pragma unroll
        for (int i = 0; i < 8; ++i) {
            int idx = i * 32 + lane;  // 0..255
            int r   = idx >> 4;       // 16 float4 per row
            int c   = idx & 15;
            const float4 v = *(const float4*)(X + (size_t)(row0 + r) * K + kc + c * 4);
            float* p = &xs[wave][r * 68 + c * 4];
            p[0] = v.x; p[1] = v.y; p[2] = v.z; p[3] = v.w;
        }
        __syncthreads();

        #pragma unroll
        for (int kk = 0; kk < 64; kk += 4) {
            const int kl = kk + 2 * khalf;
            v2f a;
            a[0] = xs[wave][lrow * 68 + kl];
            a[1] = xs[wave][lrow * 68 + kl + 1];
            const float* wr0 = W + (size_t)(kc + kl) * HID;  // row kl
            const float* wr1 = wr0 + HID;                    // row kl+1
            v2f b;
            b[0] = wr0[0 * 16 + lrow]; b[1] = wr1[0 * 16 + lrow];
            acc0 = __builtin_amdgcn_wmma_f32_16x16x4_f32(false, a, false, b, (short)0, acc0, false, false);
            b[0] = wr0[1 * 16 + lrow]; b[1] = wr1[1 * 16 + lrow];
            acc1 = __builtin_amdgcn_wmma_f32_16x16x4_f32(false, a, false, b, (short)0, acc1, false, false);
            b[0] = wr0[2 * 16 + lrow]; b[1] = wr1[2 * 16 + lrow];
            acc2 = __builtin_amdgcn_wmma_f32_16x16x4_f32(false, a, false, b, (short)0, acc2, false, false);
            b[0] = wr0[3 * 16 + lrow]; b[1] = wr1[3 * 16 + lrow];
            acc3 = __builtin_amdgcn_wmma_f32_16x16x4_f32(false, a, false, b, (short)0, acc3, false, false);
        }
        __syncthreads();
    }

    // C/D layout: VGPR j -> row = row0 + j + 8*khalf, col = lrow (+16 per n-tile)
    #pragma unroll
    for (int j = 0; j < 8; ++j) {
        float* out = H + (size_t)(row0 + 8 * khalf + j) * HID + lrow;
        out[0]  = acc0[j];
        out[16] = acc1[j];
        out[32] = acc2[j];
        out[48] = acc3[j];
    }
}

// ---------------- agg init: agg = h * dinv^2 + bias (float4, 16 threads/node) ----------------
__global__ void k_self_init(const float* __restrict__ h, const float* __restrict__ dinv,
                            const float* __restrict__ bias, float* __restrict__ agg, int n) {
    long long gid = (long long)blockIdx.x * blockDim.x + threadIdx.x;
    int i = (int)(gid >> 4);
    if (i >= n) return;
    int d4 = (int)(gid & 15);
    float di = dinv[i];
    float w  = di * di;
    const float4 hv = *(const float4*)(h + (size_t)i * HID + d4 * 4);
    const float4 bv = *(const float4*)(bias + d4 * 4);
    float4 o;
    o.x = hv.x * w + bv.x;
    o.y = hv.y * w + bv.y;
    o.z = hv.z * w + bv.z;
    o.w = hv.w * w + bv.w;
    *(float4*)(agg + (size_t)i * HID + d4 * 4) = o;
}

// ---------------- edge scatter-add: agg[dst] += h[src]*dinv[src]*dinv[dst] ----------------
// 16 threads per edge, float4 gather + 4 scalar float atomics (no packed f32 atomic on CDNA5)
__global__ void k_edge_agg(const float* __restrict__ h, const float* __restrict__ dinv,
                           const int* __restrict__ src, const int* __restrict__ dst,
                           float* __restrict__ agg, int ne) {
    long long gid = (long long)blockIdx.x * blockDim.x + threadIdx.x;
    int e = (int)(gid >> 4);
    if (e >= ne) return;
    int d4 = (int)(gid & 15);
    int s = src[e];
    int t = dst[e];
    float w = dinv[s] * dinv[t];
    const float4 hv = *(const float4*)(h + (size_t)s * HID + d4 * 4);
    float* ap = agg + (size_t)t * HID + d4 * 4;
    atomicAdd(ap + 0, hv.x * w);
    atomicAdd(ap + 1, hv.y * w);
    atomicAdd(ap + 2, hv.z * w);
    atomicAdd(ap + 3, hv.w * w);
}

// ---------------- relu in place (float4) ----------------
__global__ void k_relu_ip(float4* v, long long n4) {
    long long i = (long long)blockIdx.x * blockDim.x + threadIdx.x;
    if (i < n4) {
        float4 x = v[i];
        x.x = fmaxf(x.x, 0.0f);
        x.y = fmaxf(x.y, 0.0f);
        x.z = fmaxf(x.z, 0.0f);
        x.w = fmaxf(x.w, 0.0f);
        v[i] = x;
    }
}

// ---------------- column stats: stats[0..63]=sum, stats[64..127]=sumsq ----------------
__global__ void k_zero_stats(float* stats) {
    if (threadIdx.x < 128) stats[threadIdx.x] = 0.0f;
}

__global__ __launch_bounds__(256) void k_colstats(const float* __restrict__ v,
                                                  float* __restrict__ stats, int nrows) {
    __shared__ float ssum[256];
    __shared__ float ssq[256];
    const int col = threadIdx.x & 63;
    const int rg  = threadIdx.x >> 6;  // 0..3
    float s = 0.0f, q = 0.0f;
    for (int r = blockIdx.x * 4 + rg; r < nrows; r += gridDim.x * 4) {
        float x = v[(size_t)r * HID + col];
        s += x;
        q += x * x;
    }
    ssum[threadIdx.x] = s;
    ssq[threadIdx.x]  = q;
    __syncthreads();
    if (rg == 0) {
        s = ssum[col] + ssum[col + 64] + ssum[col + 128] + ssum[col + 192];
        q = ssq[col]  + ssq[col + 64]  + ssq[col + 128]  + ssq[col + 192];
        atomicAdd(&stats[col], s);
        atomicAdd(&stats[64 + col], q);
    }
}

// ---------------- normalize: (v - mean) / std (ddof=1), float4 ----------------
__global__ void k_normalize(float* __restrict__ v, const float* __restrict__ stats, int nrows) {
    long long gid = (long long)blockIdx.x * blockDim.x + threadIdx.x;
    int i = (int)(gid >> 4);
    if (i >= nrows) return;
    int d4 = (int)(gid & 15);
    const float n = (float)nrows;
    float* p = v + (size_t)i * HID + d4 * 4;
    float4 val = *(float4*)p;
    float vo[4] = {val.x, val.y, val.z, val.w};
    #pragma unroll
    for (int c = 0; c < 4; ++c) {
        float sum  = stats[d4 * 4 + c];
        float mean = sum / n;
        float var  = (stats[64 + d4 * 4 + c] - sum * mean) / (n - 1.0f);
        vo[c] = (vo[c] - mean) * rsqrtf(var);
    }
    val.x = vo[0]; val.y = vo[1]; val.z = vo[2]; val.w = vo[3];
    *(float4*)p = val;
}

extern "C" void kernel_launch(void* const* d_in, const int* in_sizes, int n_in,
                              void* d_out, int out_size, void* d_ws, size_t ws_size,
                              hipStream_t stream) {
    const int IN_DIM = 256;
    const int N = in_sizes[0] / IN_DIM;   // 100000
    const int E = in_sizes[1] / 2;        // 1600000

    const float* x1 = (const float*)d_in[0];
    const int*   e1 = (const int*)d_in[1];
    const float* x2 = (const float*)d_in[2];
    const int*   e2 = (const int*)d_in[3];
    const float* W1 = (const float*)d_in[4];
    const float* b1 = (const float*)d_in[5];
    const float* W2 = (const float*)d_in[6];
    const float* b2 = (const float*)d_in[7];
    float* out = (float*)d_out;

    // workspace layout (floats): dinv[N_pad] | bufH[N*64] | bufA[N*64] | stats[128]
    float* ws    = (float*)d_ws;
    const int Npad = (N + 255) & ~255;
    float* dinv  = ws;
    float* bufH  = ws + Npad;
    float* bufA  = bufH + (size_t)N * HID;
    float* stats = bufA + (size_t)N * HID;

    const int TB = 256;
    const int gN    = (N + TB - 1) / TB;
    const int gE    = (E + TB - 1) / TB;
    const int gN16  = (int)(((long long)N * 16 + TB - 1) / TB);   // 16 threads/node (float4)
    const int gE16  = (int)(((long long)E * 16 + TB - 1) / TB);   // 16 threads/edge (float4)
    const int gN4   = (int)(((long long)N * HID / 4 + TB - 1) / TB);
    const int gGemm = ((N >> 4) + 3) / 4;  // 4 row-tiles (waves) per block

    for (int g = 0; g < 2; ++g) {
        const float* X   = (g == 0) ? x1 : x2;
        const int*   src = (g == 0) ? e1 : e2;
        const int*   dst = src + E;
        float*       og  = out + (size_t)g * N * HID;

        // degree -> dinv
        k_init_ones<<<gN, TB, 0, stream>>>(dinv, N);
        k_deg_accum<<<gE, TB, 0, stream>>>(dst, dinv, E);
        k_rsqrt_ip<<<gN, TB, 0, stream>>>(dinv, N);

        // layer 1: h = X @ W1 ; agg = scatter(h) + h*dinv^2 + b1 ; relu
        k_gemm_wmma<256><<<gGemm, 128, 0, stream>>>(X, W1, bufH, N);
        k_self_init<<<gN16, TB, 0, stream>>>(bufH, dinv, b1, bufA, N);
        k_edge_agg<<<gE16, TB, 0, stream>>>(bufH, dinv, src, dst, bufA, E);
        k_relu_ip<<<gN4, TB, 0, stream>>>((float4*)bufA, (long long)N * HID / 4);

        // layer 2: h = relu_out @ W2 ; agg -> d_out segment
        k_gemm_wmma<64><<<gGemm, 128, 0, stream>>>(bufA, W2, bufH, N);
        k_self_init<<<gN16, TB, 0, stream>>>(bufH, dinv, b2, og, N);
        k_edge_agg<<<gE16, TB, 0, stream>>>(bufH, dinv, src, dst, og, E);

        // z-score in place on output segment
        k_zero_stats<<<1, 128, 0, stream>>>(stats);
        k_colstats<<<512, 256, 0, stream>>>(og, stats, N);
        k_normalize<<<gN16, TB, 0, stream>>>(og, stats, N);
    }
}